// ModelNew_48515950575840
// MI455X (gfx1250) — compile-verified
//
#include <hip/hip_runtime.h>
#include <stdint.h>

#define NROWS 8192
#define NCOLS 16384

#define TPB 256
#define CHUNK 4096                    // columns staged in LDS per pass
#define NPASS (NCOLS / CHUNK)         // 4
#define PER_THREAD (CHUNK / TPB)      // 16 elements scanned serially per thread
#define VEC_ITERS (CHUNK / (TPB * 4)) // 4 float4 moves per thread per pass
#define PAD_SHIFT 4                   // insert 4 floats of pad every 16 floats
#define PADDED_CHUNK (CHUNK + (CHUNK >> PAD_SHIFT) * 4) // 5120 floats = 20 KiB

typedef float v4f __attribute__((ext_vector_type(4)));
typedef int   v4i __attribute__((ext_vector_type(4)));

#if defined(__HIP_DEVICE_COMPILE__) &&                                         \
    __has_builtin(__builtin_amdgcn_global_load_async_to_lds_b128) &&           \
    __has_builtin(__builtin_amdgcn_s_wait_asynccnt)
#define USE_ASYNC_LDS 1
typedef __attribute__((address_space(1))) v4i as1_v4i;
typedef __attribute__((address_space(3))) v4i as3_v4i;
#else
#define USE_ASYNC_LDS 0
#endif

__device__ __forceinline__ int padded_idx(int g) {
  return g + ((g >> PAD_SHIFT) << 2);
}

__global__ __launch_bounds__(TPB) void cumprod_rows_kernel(
    const float* __restrict__ x, float* __restrict__ y) {
  __shared__ float tile[2][PADDED_CHUNK]; // double buffer, ~40 KiB total
  __shared__ float waveTot[TPB / 32];
  __shared__ float carrySh;

  const int t = threadIdx.x;
  const int lane = t & 31;
  const int wave = t >> 5;
  const long long rowBase = (long long)blockIdx.x * NCOLS;
  const float* __restrict__ grow = x + rowBase;
  float* __restrict__ growOut = y + rowBase;

  // ---- Prologue: issue async loads for chunk 0 into buffer 0 ----
#pragma unroll
  for (int i = 0; i < VEC_ITERS; ++i) {
    const int g = i * (TPB * 4) + t * 4;
    const int p = padded_idx(g);
#if USE_ASYNC_LDS
    __builtin_amdgcn_global_load_async_to_lds_b128(
        (as1_v4i*)(grow + g), (as3_v4i*)(&tile[0][p]), 0, 0);
#else
    *(v4f*)(&tile[0][p]) = *(const v4f*)(grow + g);
#endif
  }

  float seed = 1.0f;

#pragma unroll
  for (int c = 0; c < NPASS; ++c) {
    float* buf = tile[c & 1];

    // ---- Prefetch: issue next chunk's async loads into the other buffer.
    // That buffer's last readers (pass c-1 Phase C ds_loads) were drained by
    // the trailing __syncthreads() of the previous pass.
    if (c + 1 < NPASS) {
      float* nbuf = tile[(c + 1) & 1];
      const int ncb = (c + 1) * CHUNK;
#pragma unroll
      for (int i = 0; i < VEC_ITERS; ++i) {
        const int g = i * (TPB * 4) + t * 4;
        const int p = padded_idx(g);
#if USE_ASYNC_LDS
        __builtin_amdgcn_global_load_async_to_lds_b128(
            (as1_v4i*)(grow + ncb + g), (as3_v4i*)(&nbuf[p]), 0, 0);
#else
        *(v4f*)(&nbuf[p]) = *(const v4f*)(grow + ncb + g);
#endif
      }
    }

    // ASYNCcnt tracks only loads, and async loads retire in order, so
    // "<= VEC_ITERS outstanding" proves chunk c's loads have landed while
    // chunk c+1's may still be in flight.
#if USE_ASYNC_LDS
    if (c + 1 < NPASS)
      __builtin_amdgcn_s_wait_asynccnt(VEC_ITERS);
    else
      __builtin_amdgcn_s_wait_asynccnt(0);
#endif
    __syncthreads();

    // ---- Phase B1: per-thread serial product of its 16-element chunk ----
    const int base = t * (PER_THREAD + 4); // padded chunk base = t*20
    float prod = 1.0f;
#pragma unroll
    for (int j = 0; j < PER_THREAD; ++j) prod *= buf[base + j];

    // wave32 inclusive multiplicative scan of per-thread products
    float v = prod;
#pragma unroll
    for (int d = 1; d < 32; d <<= 1) {
      float n = __shfl_up(v, d, 32);
      if (lane >= d) v *= n;
    }
    if (lane == 31) waveTot[wave] = v;
    __syncthreads();

    // cross-wave exclusive prefix (8 values; broadcast LDS reads)
    float pre = seed;
    for (int w = 0; w < wave; ++w) pre *= waveTot[w];
    float ew = __shfl_up(v, 1, 32);
    float excl = (lane == 0) ? pre : pre * ew;

    // ---- Phase B2: serial rescan with seed, write prefix products in place ----
    float run = excl;
#pragma unroll
    for (int j = 0; j < PER_THREAD; ++j) {
      run *= buf[base + j];
      buf[base + j] = run;
    }
    if (t == TPB - 1) carrySh = run; // inclusive total of this chunk (with seed)
    __syncthreads();                 // orders B2 writes before Phase C reads
    seed = carrySh;

    // ---- Phase C: ds_load_b128 + non-temporal b128 store (STOREcnt path;
    // keeps ASYNCcnt loads-only and streams write-once output past L2) ----
    const int cb = c * CHUNK;
#pragma unroll
    for (int i = 0; i < VEC_ITERS; ++i) {
      const int g = i * (TPB * 4) + t * 4;
      const int p = padded_idx(g);
      __builtin_nontemporal_store(*(const v4f*)(&buf[p]),
                                  (v4f*)(growOut + cb + g));
    }

    // Drain this pass's LDS reads (DScnt) so the async engine may safely
    // overwrite this buffer when pass c+1 prefetches chunk c+2.
    __syncthreads();
  }
}

extern "C" void kernel_launch(void* const* d_in, const int* in_sizes, int n_in,
                              void* d_out, int out_size, void* d_ws,
                              size_t ws_size, hipStream_t stream) {
  (void)in_sizes; (void)n_in; (void)d_ws; (void)ws_size; (void)out_size;
  const float* x = (const float*)d_in[0];
  float* y = (float*)d_out;
  cumprod_rows_kernel<<<dim3(NROWS), dim3(TPB), 0, stream>>>(x, y);
}